// FCOSDetector_15461882266002
// MI455X (gfx1250) — compile-verified
//
#include <hip/hip_runtime.h>
#include <stdint.h>

#define NIMG 16
#define NCLS 80
#define NLOC 17064
#define PPAD 32768
#define KTOP 1000
#define KPAD 1008
#define NT   63      /* 63*16 = 1008 tiles along each NMS axis */
#define MASKW 64     /* halfwords per mask row */

typedef float v2f __attribute__((ext_vector_type(2)));
typedef float v8f __attribute__((ext_vector_type(8)));
typedef unsigned long long u64;
typedef unsigned int u32;
typedef unsigned short u16;

struct LevelPtrs {
  const float* cls[5];
  const float* reg[5];
  const float* ctr[5];
};

__device__ __forceinline__ float sigm(float x) { return 1.0f / (1.0f + expf(-x)); }

// ---------------- Phase 1: score + argmax + box decode + key pack ----------------
__global__ __launch_bounds__(256)
void k_score(LevelPtrs in, u64* __restrict__ keys, float4* __restrict__ boxes,
             int* __restrict__ labels) {
  int t = blockIdx.x * blockDim.x + threadIdx.x;
  if (t >= NIMG * PPAD) return;
  int n = t >> 15;
  int s = t & (PPAD - 1);
  if (s >= NLOC) { keys[t] = 0ull; return; }

  int l, off, w, stride;
  if      (s < 12800) { l = 0; off = 0;     w = 128; stride = 8;   }
  else if (s < 16000) { l = 1; off = 12800; w = 64;  stride = 16;  }
  else if (s < 16800) { l = 2; off = 16000; w = 32;  stride = 32;  }
  else if (s < 17008) { l = 3; off = 16800; w = 16;  stride = 64;  }
  else                { l = 4; off = 17008; w = 8;   stride = 128; }
  static const int HWs[5] = {12800, 3200, 800, 208, 56};
  size_t HW = (size_t)HWs[l];

  int hw = s - off;
  int y = hw / w, x = hw - y * w;

  const float* cb = in.cls[l] + (size_t)n * NCLS * HW + hw;
  float best = -3.402823466e38f; int bi = 0;
  for (int c = 0; c < NCLS; ++c) {
    float v = cb[(size_t)c * HW];
    if (v > best) { best = v; bi = c; }   // first-occurrence argmax, matches jnp.argmax
  }
  float ctr = in.ctr[l][(size_t)n * HW + hw];
  float sc = sqrtf(sigm(best) * sigm(ctr));
  if (!(sc > 0.05f)) sc = 0.0f;

  // key: score desc, then smaller loc first (matches lax.top_k tie-break)
  keys[t] = ((u64)__float_as_uint(sc) << 32) | (u64)(0xFFFFFFFFu - (u32)s);

  const float* rb = in.reg[l] + (size_t)n * 4 * HW + hw;
  float r0 = rb[0], r1 = rb[HW], r2 = rb[2 * HW], r3 = rb[3 * HW];
  float px = (float)x * (float)stride + 0.5f * (float)stride;
  float py = (float)y * (float)stride + 0.5f * (float)stride;
  float4 bx;
  bx.x = fminf(fmaxf(px - r0, 0.f), 1024.f);
  bx.y = fminf(fmaxf(py - r1, 0.f), 800.f);
  bx.z = fminf(fmaxf(px + r2, 0.f), 1024.f);
  bx.w = fminf(fmaxf(py + r3, 0.f), 800.f);
  boxes [(size_t)n * NLOC + s] = bx;
  labels[(size_t)n * NLOC + s] = bi;
}

// ---------------- Phase 2: per-image bitonic sort, 256KB resident in LDS ----------------
__global__ __launch_bounds__(1024)
void k_sort(const u64* __restrict__ keys, u64* __restrict__ topk) {
  extern __shared__ u64 sm[];                     // 32768 * 8B = 256KB (<= 320KB/WGP)
  int n = blockIdx.x, t = threadIdx.x;
  for (int i = t; i < PPAD; i += 1024) sm[i] = keys[(size_t)n * PPAD + i];
  __syncthreads();
  for (int k = 2; k <= PPAD; k <<= 1) {
    for (int j = k >> 1; j > 0; j >>= 1) {
      for (int i = t; i < PPAD; i += 1024) {
        int ixj = i ^ j;
        if (ixj > i) {
          u64 a = sm[i], b = sm[ixj];
          bool desc = ((i & k) == 0);
          if (desc ? (a < b) : (a > b)) { sm[i] = b; sm[ixj] = a; }
        }
      }
      __syncthreads();
    }
  }
  if (t < KPAD) topk[n * KPAD + t] = (t < KTOP) ? sm[t] : 0ull;
}

// ---------------- Phase 3: gather top-1000 + class-offset boxes + areas ----------------
__global__ __launch_bounds__(256)
void k_gather(const u64* __restrict__ topk, const float4* __restrict__ boxes,
              const int* __restrict__ labels, float* __restrict__ tscore,
              float4* __restrict__ tbox, float4* __restrict__ tob,
              int* __restrict__ tlabel, float* __restrict__ tarea) {
  int t = blockIdx.x * blockDim.x + threadIdx.x;
  if (t >= NIMG * KPAD) return;
  int n = t / KPAD, k = t - n * KPAD;
  if (k < KTOP) {
    u64 key = topk[t];
    float sc = __uint_as_float((u32)(key >> 32));
    u32 loc = 0xFFFFFFFFu - (u32)(key & 0xFFFFFFFFull);
    float4 bx = boxes[(size_t)n * NLOC + loc];
    int lb = labels[(size_t)n * NLOC + loc];
    float ofs = (float)lb * 4096.0f;
    float4 ob = make_float4(bx.x + ofs, bx.y + ofs, bx.z + ofs, bx.w + ofs);
    tscore[t] = sc; tbox[t] = bx; tob[t] = ob; tlabel[t] = lb;
    tarea[t] = (ob.z - ob.x) * (ob.w - ob.y);
  } else {
    tscore[t] = 0.f;
    tbox[t] = make_float4(0.f, 0.f, 0.f, 0.f);
    tob[t]  = make_float4(0.f, 0.f, 0.f, 0.f);
    tlabel[t] = 0; tarea[t] = 0.f;
  }
}

// ---------------- Phase 4a: 16x16 IoU tiles; area_i + area_j via V_WMMA_F32_16X16X4_F32 ----
__global__ __launch_bounds__(32)
void k_iou(const float4* __restrict__ tob, const float* __restrict__ tarea,
           u16* __restrict__ mask) {
  int b = blockIdx.x;
  int n = b / (NT * NT);
  int r = b - n * NT * NT;
  int ti = r / NT, tj = r - ti * NT;
  int lane = threadIdx.x;

  __shared__ float4 bi[16], bj[16];
  if (lane < 16) bi[lane]      = tob[n * KPAD + ti * 16 + lane];
  else           bj[lane - 16] = tob[n * KPAD + tj * 16 + (lane - 16)];

  float ai = 0.f, aj = 0.f;
  if (lane < 16) {
    ai = tarea[n * KPAD + ti * 16 + lane];
    aj = tarea[n * KPAD + tj * 16 + lane];
  }
  // A (16x4 f32): VGPR0 = K0(l0-15)/K2(l16-31), VGPR1 = K1/K3. Row m = lane.
  v2f A; A[0] = (lane < 16) ? ai  : 0.f; A[1] = (lane < 16) ? 1.f : 0.f;
  // B (4x16 f32): VGPR0 = row K0(l0-15)/K2(l16-31), VGPR1 = K1/K3. Col n = lane&15.
  v2f B; B[0] = (lane < 16) ? 1.f : 0.f; B[1] = (lane < 16) ? aj  : 0.f;
  v8f C = {0.f, 0.f, 0.f, 0.f, 0.f, 0.f, 0.f, 0.f};
  // D[m][n] = area_i[m] + area_j[n]
  v8f D = __builtin_amdgcn_wmma_f32_16x16x4_f32(false, A, false, B, (short)0, C,
                                                false, false);
  __syncthreads();

  int nn = lane & 15;
  int mo = (lane >> 4) * 8;          // D VGPR rr: lanes0-15 -> M=rr, lanes16-31 -> M=rr+8
  float4 Bj = bj[nn];
  for (int rr = 0; rr < 8; ++rr) {
    float4 Bi = bi[rr + mo];
    float ww = fmaxf(fminf(Bi.z, Bj.z) - fmaxf(Bi.x, Bj.x), 0.f);
    float hh = fmaxf(fminf(Bi.w, Bj.w) - fmaxf(Bi.y, Bj.y), 0.f);
    float inter = ww * hh;
    float uni = D[rr] - inter + 1e-9f;
    bool supp = (inter / uni) > 0.6f;
    u32 bal = (u32)__ballot(supp);   // wave32: bits0-15 = row rr, bits16-31 = row rr+8
    int row0 = n * KPAD + ti * 16 + rr;
    if (lane == 0) mask[(size_t)row0 * MASKW + tj]        = (u16)(bal & 0xFFFFu);
    if (lane == 1) mask[(size_t)(row0 + 8) * MASKW + tj]  = (u16)(bal >> 16);
  }
}

// ---------------- Phase 4b: greedy NMS scan + top-100 selection ----------------
__global__ __launch_bounds__(1024)
void k_nms(const float* __restrict__ tscore, const float4* __restrict__ tbox,
           const int* __restrict__ tlabel, const u16* __restrict__ mask,
           float* __restrict__ out) {
  __shared__ unsigned char valid[1024];
  __shared__ int ranks[1024];
  int n = blockIdx.x, t = threadIdx.x;
  valid[t] = (t < KTOP) && (tscore[n * KPAD + t] > 0.f);
  __syncthreads();
  for (int i = 0; i < KTOP; ++i) {
    if (t > i && t < KTOP && valid[i]) {
      u16 m = mask[(size_t)(n * KPAD + i) * MASKW + (t >> 4)];
      if ((m >> (t & 15)) & 1) valid[t] = 0;
    }
    __syncthreads();
  }
  if (t == 0) {
    int cnt = 0;
    for (int p = 0; p < KTOP; ++p) if (valid[p])  ranks[p] = cnt++;
    for (int p = 0; p < KTOP; ++p) if (!valid[p]) ranks[p] = cnt++;
  }
  __syncthreads();
  if (t < KTOP) {
    int rk = ranks[t];
    if (rk < 100) {
      float4 bx = tbox[n * KPAD + t];
      float* ob = out + ((size_t)n * 100 + rk) * 4;
      ob[0] = bx.x; ob[1] = bx.y; ob[2] = bx.z; ob[3] = bx.w;
      out[6400 + n * 100 + rk] = valid[t] ? tscore[n * KPAD + t] : 0.f;
      out[8000 + n * 100 + rk] = (float)tlabel[n * KPAD + t];
    }
  }
}

extern "C" void kernel_launch(void* const* d_in, const int* in_sizes, int n_in,
                              void* d_out, int out_size, void* d_ws, size_t ws_size,
                              hipStream_t stream) {
  (void)in_sizes; (void)n_in; (void)out_size; (void)ws_size;
  LevelPtrs in;
  for (int l = 0; l < 5; ++l) {
    in.cls[l] = (const float*)d_in[3 * l + 0];
    in.reg[l] = (const float*)d_in[3 * l + 1];
    in.ctr[l] = (const float*)d_in[3 * l + 2];
  }
  char* ws = (char*)d_ws;
  u64*    keys   = (u64*)   (ws + 0);          // 16*32768*8    = 4,194,304
  float4* boxes  = (float4*)(ws + 4194304);    // 16*17064*16   = 4,368,384
  int*    labels = (int*)   (ws + 8562688);    // 16*17064*4    = 1,092,096
  u64*    topk   = (u64*)   (ws + 9654784);    // 16*1008*8     =   129,024
  float*  tscore = (float*) (ws + 9783808);    // 16*1008*4     =    64,512
  float4* tbox   = (float4*)(ws + 9848320);    // 16*1008*16    =   258,048
  float4* tob    = (float4*)(ws + 10106368);   // 16*1008*16    =   258,048
  int*    tlabel = (int*)   (ws + 10364416);   // 16*1008*4     =    64,512
  float*  tarea  = (float*) (ws + 10428928);   // 16*1008*4     =    64,512
  u16*    mask   = (u16*)   (ws + 10493440);   // 16*1008*64*2  = 2,064,384  -> end 12,557,824

  k_score <<<(NIMG * PPAD + 255) / 256, 256, 0, stream>>>(in, keys, boxes, labels);
  k_sort  <<<NIMG, 1024, PPAD * sizeof(u64), stream>>>(keys, topk);
  k_gather<<<(NIMG * KPAD + 255) / 256, 256, 0, stream>>>(topk, boxes, labels,
                                                          tscore, tbox, tob, tlabel, tarea);
  k_iou   <<<NIMG * NT * NT, 32, 0, stream>>>(tob, tarea, mask);
  k_nms   <<<NIMG, 1024, 0, stream>>>(tscore, tbox, tlabel, mask, (float*)d_out);
}